// clustering_attention_45286135169491
// MI455X (gfx1250) — compile-verified
//
#include <hip/hip_runtime.h>

// Problem constants (from setup_inputs)
constexpr int B  = 16;
constexpr int N  = 2048;    // nodes per batch
constexpr int S  = 12;      // feature dim
constexpr int SO = 12;      // projected dim
constexpr int K  = 32;      // neighbors per node == wave32
#define ALPHA 0.5f

typedef __attribute__((ext_vector_type(2))) float v2f;
typedef __attribute__((ext_vector_type(8))) float v8f;

// -----------------------------------------------------------------------------
// Kernel 1: e_src/e_dst per node via V_WMMA_F32_16X16X4_F32.
//   e_src[n] = input[n,:] . (W @ a[:SO]) ; e_dst[n] = input[n,:] . (W @ a[SO:])
// Each wave processes a 16-row tile: A = 16x4 slab of input, B holds the two
// projection vectors in columns 0/1 (built branchlessly so EXEC stays all-1s
// through the WMMA chain), accumulating 3 chained K=4 WMMAs (S = 12 = 3*4).
// -----------------------------------------------------------------------------
__global__ __launch_bounds__(256) void gat_e_wmma(
    const float* __restrict__ X,   // (B*N, S)
    const float* __restrict__ Wm,  // (S, SO)
    const float* __restrict__ av,  // (2*SO)
    float* __restrict__ esrc,      // (B*N)
    float* __restrict__ edst)      // (B*N)
{
    __shared__ float wsrc[S], wdst[S];
    const int t = threadIdx.x;
    if (t < 2 * S) {
        const int s = (t < S) ? t : t - S;
        const float* ap = av + ((t < S) ? 0 : SO);
        float acc = 0.f;
#pragma unroll
        for (int j = 0; j < SO; ++j) acc = fmaf(Wm[s * SO + j], ap[j], acc);
        if (t < S) wsrc[s] = acc; else wdst[s] = acc;
    }
    __syncthreads();

    const int lane = t & 31;
    const int wave = t >> 5;
    const int tile = blockIdx.x * 8 + wave;   // 16 rows per tile, 8 waves/block
    const int row0 = tile * 16;
    const int m    = lane & 15;               // A: row / B: column (lane dim)
    const int kb   = (lane >> 4) << 1;        // lanes 16-31 hold K+2,K+3

    // Branchless column select: col 0 = wsrc, col 1 = wdst, cols 2-15 = 0.
    const float c0 = (m == 0) ? 1.f : 0.f;
    const float c1 = (m == 1) ? 1.f : 0.f;

    const float* arow = X + (size_t)(row0 + m) * S;
    v8f c = {0.f, 0.f, 0.f, 0.f, 0.f, 0.f, 0.f, 0.f};
#pragma unroll
    for (int k0 = 0; k0 < S; k0 += 4) {
        v2f a, b;
        a.x = arow[k0 + kb];
        a.y = arow[k0 + kb + 1];
        b.x = fmaf(c0, wsrc[k0 + kb],     c1 * wdst[k0 + kb]);
        b.y = fmaf(c0, wsrc[k0 + kb + 1], c1 * wdst[k0 + kb + 1]);
        // (neg_a, A, neg_b, B, c_mod, C, reuse_a, reuse_b)
        c = __builtin_amdgcn_wmma_f32_16x16x4_f32(false, a, false, b,
                                                  (short)0, c, false, false);
    }

    // D layout: VGPR r, lanes 0-15 -> (M=r, N=lane); lanes 16-31 -> (M=r+8).
    // Column 0 = e_src (lanes 0 & 16), column 1 = e_dst (lanes 1 & 17).
    const int rbase = row0 + ((lane >> 4) << 3);
    if (m == 0) {
#pragma unroll
        for (int r = 0; r < 8; ++r) esrc[rbase + r] = c[r];
    } else if (m == 1) {
#pragma unroll
        for (int r = 0; r < 8; ++r) edst[rbase + r] = c[r];
    }
}

// -----------------------------------------------------------------------------
// Kernel 2: one wave32 per node; lane k owns neighbor k (K == 32).
// Gather e_dst, leaky_relu, wave-reduction softmax, scale gathered neighbor
// features, fully-coalesced 1536B store per wave. Store-bandwidth bound:
// ~50.3 MB output + ~29 MB L2-resident reads -> ~3.5 us at 23.3 TB/s.
// -----------------------------------------------------------------------------
__global__ __launch_bounds__(256) void gat_gather_softmax(
    const float* __restrict__ X,    // (B*N, S)
    const int*   __restrict__ idx,  // (B*N, K)
    const float* __restrict__ esrc,
    const float* __restrict__ edst,
    float* __restrict__ out)        // (B*N, K, S)
{
    const int lane = threadIdx.x & 31;
    const int wave = threadIdx.x >> 5;
    const int bn   = blockIdx.x * 8 + wave;   // flat node id
    const int b    = bn >> 11;                // / N (N = 2048)

    const int j = idx[(size_t)bn * K + lane];
    const float e  = esrc[bn] + edst[(b << 11) + j];
    const float sc = (e > 0.f) ? e : ALPHA * e;   // leaky_relu, slope 0.5

    float mx = sc;
#pragma unroll
    for (int off = 16; off > 0; off >>= 1) mx = fmaxf(mx, __shfl_xor(mx, off, 32));
    const float p = __expf(sc - mx);
    float sum = p;
#pragma unroll
    for (int off = 16; off > 0; off >>= 1) sum += __shfl_xor(sum, off, 32);
    const float att = p / sum;

    const float4* row = (const float4*)(X + ((size_t)(b << 11) + j) * S); // 48B rows, 16B aligned
    float4 r0 = row[0], r1 = row[1], r2 = row[2];
    r0.x *= att; r0.y *= att; r0.z *= att; r0.w *= att;
    r1.x *= att; r1.y *= att; r1.z *= att; r1.w *= att;
    r2.x *= att; r2.y *= att; r2.z *= att; r2.w *= att;

    float4* o = (float4*)(out + (size_t)bn * (K * S) + lane * S);
    o[0] = r0; o[1] = r1; o[2] = r2;
}

// -----------------------------------------------------------------------------
// Inputs (setup_inputs order): 0 fushed_features (UNUSED by reference),
// 1 input_data, 2 W, 3 a, 4 idx. Output: (B, N, K, S) float32.
// -----------------------------------------------------------------------------
extern "C" void kernel_launch(void* const* d_in, const int* in_sizes, int n_in,
                              void* d_out, int out_size, void* d_ws, size_t ws_size,
                              hipStream_t stream) {
    (void)in_sizes; (void)n_in; (void)out_size; (void)ws_size;
    const float* X   = (const float*)d_in[1];
    const float* Wm  = (const float*)d_in[2];
    const float* av  = (const float*)d_in[3];
    const int*   idx = (const int*)d_in[4];
    float* out  = (float*)d_out;
    float* esrc = (float*)d_ws;                 // B*N floats
    float* edst = esrc + (size_t)B * N;         // B*N floats (total 256 KB)

    // 2048 tiles of 16 rows, 8 waves per 256-thread block -> 256 blocks
    gat_e_wmma<<<(B * N / 16) / 8, 256, 0, stream>>>(X, Wm, av, esrc, edst);
    // 32768 nodes, 8 waves per block -> 4096 blocks
    gat_gather_softmax<<<(B * N) / 8, 256, 0, stream>>>(X, idx, esrc, edst, out);
}